// Generator_LSTM_23433341567859
// MI455X (gfx1250) — compile-verified
//
#include <hip/hip_runtime.h>
#include <hip/hip_bf16.h>

// ---------------------------------------------------------------------------
// CDNA5 (gfx1250) stacked-LSTM generator, V_WMMA_F32_16X16X4_F32 recurrence.
// z is computed transposed (gates x batch, 16 batch columns per wave).
// Gate rows of the A-operand (W^T/U^T) are PERMUTED so that each lane half
// receives exactly the gates of the state channels it owns:
//   half0 owns h/c channels {0,1,4,5}, half1 owns {2,3,6,7}
// which matches the h^T B-operand layout exactly -> the recurrence needs
// NO cross-lane shuffles and NO half-selects at all.
// Activations use v_exp_f32 + v_rcp_f32 (no IEEE division chains).
// BatchNorm (training mode) via wave reductions + f32 global atomics.
// ---------------------------------------------------------------------------

typedef float v2f __attribute__((ext_vector_type(2)));
typedef float v8f __attribute__((ext_vector_type(8)));

__device__ __forceinline__ v8f wmma4(v2f a, v2f b, v8f c) {
  // D = A(16x4 f32) * B(4x16 f32) + C(16x16 f32)
  return __builtin_amdgcn_wmma_f32_16x16x4_f32(false, a, false, b,
                                               (short)0, c, false, false);
}

// sigmoid via v_exp_f32 + v_rcp_f32 (1-ulp rcp, no div-scale/fixup chain)
__device__ __forceinline__ float fsig(float x) {
  return __builtin_amdgcn_rcpf(1.0f + __expf(-x));
}
// tanh(x) = 1 - 2/(exp(2x)+1), same fast path
__device__ __forceinline__ float ftanh(float x) {
  return 1.0f - 2.0f * __builtin_amdgcn_rcpf(__expf(2.0f * x) + 1.0f);
}
// sum over the 16 lanes of each half-wave (used once per kernel, not per step)
__device__ __forceinline__ float rsum16(float v) {
  v += __shfl_xor(v, 1, 32);
  v += __shfl_xor(v, 2, 32);
  v += __shfl_xor(v, 4, 32);
  v += __shfl_xor(v, 8, 32);
  return v;
}

#define EPS 1e-3f
#define T_ 32
#define F_ 16
#define RW 14

// ---------------------------------------------------------------- zero stats
__global__ void k0_zero(float* p, int n) {
  int t = threadIdx.x;
  if (t < n) p[t] = 0.0f;
}

// ------------------------------------------------- K1: LSTM0 (tanh), T=32
__global__ void k1_lstm0(const float* __restrict__ X, const float* __restrict__ W0,
                         const float* __restrict__ U0, const float* __restrict__ b0,
                         float* __restrict__ hbuf, float* __restrict__ cbuf,
                         float* __restrict__ stats0) {
  const int lane = threadIdx.x & 31;
  const int hf = lane >> 4;        // lane half (K-offset for A/B operands)
  const int l = lane & 15;         // batch column within wave / A-matrix M row
  const int wave = (blockIdx.x * blockDim.x + threadIdx.x) >> 5;
  const size_t b = (size_t)wave * 16 + l;

  // Permuted A rows: row M=l of tile m is z-column
  //   zc = (q&3)*8 + 4m + 2*(l>>3) + (q>>2),  q = l&7  (gates i,f,g,o stride 8)
  const int q_l = l & 7;
  const int zcb = (q_l & 3) * 8 + 2 * (l >> 3) + (q_l >> 2);  // + 4*m

  v2f Wt[2][4], Ut[2][2];
#pragma unroll
  for (int m = 0; m < 2; m++) {
    const int zc = zcb + 4 * m;
#pragma unroll
    for (int c = 0; c < 4; c++) {
      int kg = 4 * c + 2 * hf;
      Wt[m][c].x = W0[kg * 32 + zc];
      Wt[m][c].y = W0[(kg + 1) * 32 + zc];
    }
#pragma unroll
    for (int c = 0; c < 2; c++) {
      int kg = 4 * c + 2 * hf;
      Ut[m][c].x = U0[kg * 32 + zc];
      Ut[m][c].y = U0[(kg + 1) * 32 + zc];
    }
  }
  // Bias in permuted D layout: element q of tile m (this lane's half) is
  //   zc = (q&3)*8 + 4m + 2*hf + (q>>2)
  float bias[2][8];
#pragma unroll
  for (int m = 0; m < 2; m++)
#pragma unroll
    for (int q = 0; q < 8; q++)
      bias[m][q] = b0[(q & 3) * 8 + 4 * m + 2 * hf + (q >> 2)];

  // Owned state: slot s -> channel r = 4*(s>>1) + 2*hf + (s&1)
  float hr[4], cr[4];
#pragma unroll
  for (int s = 0; s < 4; s++) { hr[s] = 0.0f; cr[s] = 0.0f; }

  const float* xrow = X + b * (size_t)(T_ * F_) + 2 * hf;
  for (int t = 0; t < T_; t++) {
    v2f xT[4];
#pragma unroll
    for (int c = 0; c < 4; c++)
      xT[c] = *(const v2f*)(xrow + t * F_ + 4 * c);
    v2f hT[2];
#pragma unroll
    for (int c = 0; c < 2; c++) { hT[c].x = hr[2 * c]; hT[c].y = hr[2 * c + 1]; }

    v8f acc[2];
#pragma unroll
    for (int m = 0; m < 2; m++)
#pragma unroll
      for (int q = 0; q < 8; q++) acc[m][q] = bias[m][q];
#pragma unroll
    for (int c = 0; c < 4; c++) {
      acc[0] = wmma4(Wt[0][c], xT[c], acc[0]);
      acc[1] = wmma4(Wt[1][c], xT[c], acc[1]);
    }
#pragma unroll
    for (int c = 0; c < 2; c++) {
      acc[0] = wmma4(Ut[0][c], hT[c], acc[0]);
      acc[1] = wmma4(Ut[1][c], hT[c], acc[1]);
    }
    // Gates arrive pre-sorted per lane: [i, f, g, o] per owned channel.
#pragma unroll
    for (int m = 0; m < 2; m++)
#pragma unroll
      for (int p = 0; p < 2; p++) {
        int s = 2 * m + p;
        float iv = acc[m][4 * p + 0];
        float fv = acc[m][4 * p + 1];
        float gv = acc[m][4 * p + 2];
        float ov = acc[m][4 * p + 3];
        float cn = fsig(fv) * cr[s] + fsig(iv) * ftanh(gv);
        cr[s] = cn;
        hr[s] = fsig(ov) * ftanh(cn);
      }
  }
  // Each half stores its owned channels {2hf,2hf+1} and {4+2hf,5+2hf}.
  *(v2f*)(hbuf + b * 8 + 2 * hf)     = (v2f){hr[0], hr[1]};
  *(v2f*)(hbuf + b * 8 + 4 + 2 * hf) = (v2f){hr[2], hr[3]};
  *(v2f*)(cbuf + b * 8 + 2 * hf)     = (v2f){cr[0], cr[1]};
  *(v2f*)(cbuf + b * 8 + 4 + 2 * hf) = (v2f){cr[2], cr[3]};

  // BN partial sums: each half reduces its owned channels over 16 batch lanes.
#pragma unroll
  for (int s = 0; s < 4; s++) {
    int r = 4 * (s >> 1) + 2 * hf + (s & 1);
    float sh = rsum16(hr[s]);
    float qh = rsum16(hr[s] * hr[s]);
    float sc = rsum16(cr[s]);
    float qc = rsum16(cr[s] * cr[s]);
    if (l == 0) {
      atomicAdd(&stats0[r], sh);
      atomicAdd(&stats0[8 + r], qh);
      atomicAdd(&stats0[16 + r], sc);
      atomicAdd(&stats0[24 + r], qc);
    }
  }
}

// --------------------- K2: finalize BN(h), BN(c); V1 = W1 + U1
__global__ void k2_fin0(const float* __restrict__ stats0,
                        const float* __restrict__ gh, const float* __restrict__ bh,
                        const float* __restrict__ gc, const float* __restrict__ bc,
                        const float* __restrict__ W1, const float* __restrict__ U1,
                        float* __restrict__ aff, float* __restrict__ V1, float invB) {
  int t = threadIdx.x;
  if (t < 8) {
    float m = stats0[t] * invB;
    float v = stats0[8 + t] * invB - m * m;
    float s = gh[t] * rsqrtf(v + EPS);
    aff[t] = s;
    aff[8 + t] = bh[t] - m * s;
    float mc = stats0[16 + t] * invB;
    float vc = stats0[24 + t] * invB - mc * mc;
    float sc = gc[t] * rsqrtf(vc + EPS);
    aff[16 + t] = sc;
    aff[24 + t] = bc[t] - mc * sc;
  }
  if (t < 256) V1[t] = W1[t] + U1[t];
}

// ---------------- K3: autoregressive LSTM1 (relu), 14 steps, x = h
__global__ void k3_lstm1(const float* __restrict__ hbuf, const float* __restrict__ cbuf,
                         const float* __restrict__ aff, const float* __restrict__ V1,
                         const float* __restrict__ b1,
                         float* __restrict__ bars, float* __restrict__ stats3) {
  const int lane = threadIdx.x & 31;
  const int hf = lane >> 4;
  const int l = lane & 15;
  const int wave = (blockIdx.x * blockDim.x + threadIdx.x) >> 5;
  const size_t b = (size_t)wave * 16 + l;

  const int q_l = l & 7;
  const int zcb = (q_l & 3) * 8 + 2 * (l >> 3) + (q_l >> 2);

  v2f Vt[2][2];
#pragma unroll
  for (int m = 0; m < 2; m++) {
    const int zc = zcb + 4 * m;
#pragma unroll
    for (int c = 0; c < 2; c++) {
      int kg = 4 * c + 2 * hf;
      Vt[m][c].x = V1[kg * 32 + zc];
      Vt[m][c].y = V1[(kg + 1) * 32 + zc];
    }
  }
  float bias[2][8];
#pragma unroll
  for (int m = 0; m < 2; m++)
#pragma unroll
    for (int q = 0; q < 8; q++)
      bias[m][q] = b1[(q & 3) * 8 + 4 * m + 2 * hf + (q >> 2)];

  // Load owned channels with BN affine applied.
  float hr[4], cr[4], s3[4], q3[4];
  {
    v2f h01 = *(const v2f*)(hbuf + b * 8 + 2 * hf);
    v2f h23 = *(const v2f*)(hbuf + b * 8 + 4 + 2 * hf);
    v2f c01 = *(const v2f*)(cbuf + b * 8 + 2 * hf);
    v2f c23 = *(const v2f*)(cbuf + b * 8 + 4 + 2 * hf);
    float hv[4] = {h01.x, h01.y, h23.x, h23.y};
    float cv[4] = {c01.x, c01.y, c23.x, c23.y};
#pragma unroll
    for (int s = 0; s < 4; s++) {
      int r = 4 * (s >> 1) + 2 * hf + (s & 1);
      hr[s] = aff[r] * hv[s] + aff[8 + r];
      cr[s] = aff[16 + r] * cv[s] + aff[24 + r];
      s3[s] = 0.0f;
      q3[s] = 0.0f;
    }
  }

  for (int t = 0; t < RW; t++) {
    v2f hT[2];
#pragma unroll
    for (int c = 0; c < 2; c++) { hT[c].x = hr[2 * c]; hT[c].y = hr[2 * c + 1]; }
    v8f acc[2];
#pragma unroll
    for (int m = 0; m < 2; m++)
#pragma unroll
      for (int q = 0; q < 8; q++) acc[m][q] = bias[m][q];
#pragma unroll
    for (int c = 0; c < 2; c++) {
      acc[0] = wmma4(Vt[0][c], hT[c], acc[0]);
      acc[1] = wmma4(Vt[1][c], hT[c], acc[1]);
    }
#pragma unroll
    for (int m = 0; m < 2; m++)
#pragma unroll
      for (int p = 0; p < 2; p++) {
        int s = 2 * m + p;
        float iv = acc[m][4 * p + 0];
        float fv = acc[m][4 * p + 1];
        float gv = acc[m][4 * p + 2];
        float ov = acc[m][4 * p + 3];
        float cn = fsig(fv) * cr[s] + fsig(iv) * fmaxf(gv, 0.0f);
        cr[s] = cn;
        hr[s] = fsig(ov) * fmaxf(cn, 0.0f);
        s3[s] += hr[s];
        q3[s] += hr[s] * hr[s];
      }
    // bars reversed: row = 13 - t; each half stores its owned channels.
    float* dst = bars + ((size_t)b * RW + (RW - 1 - t)) * 8;
    *(v2f*)(dst + 2 * hf)     = (v2f){hr[0], hr[1]};
    *(v2f*)(dst + 4 + 2 * hf) = (v2f){hr[2], hr[3]};
  }
#pragma unroll
  for (int s = 0; s < 4; s++) {
    int r = 4 * (s >> 1) + 2 * hf + (s & 1);
    float sv = rsum16(s3[s]);
    float qv = rsum16(q3[s]);
    if (l == 0) {
      atomicAdd(&stats3[r], sv);
      atomicAdd(&stats3[8 + r], qv);
    }
  }
}

// ------------------------ K4: finalize BN3 (over B*ROWS)
__global__ void k4_fin3(const float* __restrict__ stats3,
                        const float* __restrict__ g3, const float* __restrict__ b3,
                        float* __restrict__ aff, float invN) {
  int t = threadIdx.x;
  if (t < 8) {
    float m = stats3[t] * invN;
    float v = stats3[8 + t] * invN - m * m;
    float s = g3[t] * rsqrtf(v + EPS);
    aff[32 + t] = s;
    aff[40 + t] = b3[t] - m * s;
  }
}

// --------------- K5: backward LSTM2 (relu), return_sequences, D=8 -> Do=4
__global__ void k5_lstm2(const float* __restrict__ bars, const float* __restrict__ aff,
                         const float* __restrict__ W2, const float* __restrict__ U2,
                         const float* __restrict__ b2, float* __restrict__ out) {
  const int lane = threadIdx.x & 31;
  const int hf = lane >> 4;
  const int l = lane & 15;
  const int wave = (blockIdx.x * blockDim.x + threadIdx.x) >> 5;
  const size_t b = (size_t)wave * 16 + l;

  // Permuted rows, gate stride 4 (z width 16): zc = (q&3)*4 + 2*(l>>3) + (q>>2)
  const int q_l = l & 7;
  const int zc = (q_l & 3) * 4 + 2 * (l >> 3) + (q_l >> 2);

  v2f W2t[2], U2t;
#pragma unroll
  for (int c = 0; c < 2; c++) {
    int kg = 4 * c + 2 * hf;
    W2t[c].x = W2[kg * 16 + zc];
    W2t[c].y = W2[(kg + 1) * 16 + zc];
  }
  {
    int kg = 2 * hf;
    U2t.x = U2[kg * 16 + zc];
    U2t.y = U2[(kg + 1) * 16 + zc];
  }
  float bias[8];
#pragma unroll
  for (int q = 0; q < 8; q++) bias[q] = b2[(q & 3) * 4 + 2 * hf + (q >> 2)];

  // Owned state: slot p -> channel r = 2*hf + p (half0: {0,1}, half1: {2,3})
  float hr[2], cr[2];
#pragma unroll
  for (int p = 0; p < 2; p++) { hr[p] = 0.0f; cr[p] = 0.0f; }

  for (int t = 0; t < RW; t++) {
    float xr[8];
#pragma unroll
    for (int j = 0; j < 8; j++)  // BN3 affine applied on load
      xr[j] = aff[32 + j] * bars[((size_t)b * RW + t) * 8 + j] + aff[40 + j];
    v2f xT[2];
#pragma unroll
    for (int c = 0; c < 2; c++) {
      xT[c].x = xr[4 * c + 2 * hf];
      xT[c].y = xr[4 * c + 2 * hf + 1];
    }
    v2f hT;
    hT.x = hr[0];
    hT.y = hr[1];

    v8f a;
#pragma unroll
    for (int q = 0; q < 8; q++) a[q] = bias[q];
    a = wmma4(W2t[0], xT[0], a);
    a = wmma4(W2t[1], xT[1], a);
    a = wmma4(U2t, hT, a);

#pragma unroll
    for (int p = 0; p < 2; p++) {
      float iv = a[4 * p + 0];
      float fv = a[4 * p + 1];
      float gv = a[4 * p + 2];
      float ov = a[4 * p + 3];
      float cn = fsig(fv) * cr[p] + fsig(iv) * fmaxf(gv, 0.0f);
      cr[p] = cn;
      hr[p] = fsig(ov) * fmaxf(cn, 0.0f);
    }
    *(v2f*)(out + ((size_t)b * RW + t) * 4 + 2 * hf) = (v2f){hr[0], hr[1]};
  }
}

// ---------------------------------------------------------------------------
extern "C" void kernel_launch(void* const* d_in, const int* in_sizes, int n_in,
                              void* d_out, int out_size, void* d_ws, size_t ws_size,
                              hipStream_t stream) {
  const float* X  = (const float*)d_in[0];
  const float* W0 = (const float*)d_in[1];
  const float* U0 = (const float*)d_in[2];
  const float* b0 = (const float*)d_in[3];
  const float* gh = (const float*)d_in[4];
  const float* bh = (const float*)d_in[5];
  const float* gc = (const float*)d_in[6];
  const float* bc = (const float*)d_in[7];
  const float* W1 = (const float*)d_in[8];
  const float* U1 = (const float*)d_in[9];
  const float* b1 = (const float*)d_in[10];
  const float* g3 = (const float*)d_in[11];
  const float* b3 = (const float*)d_in[12];
  const float* W2 = (const float*)d_in[13];
  const float* U2 = (const float*)d_in[14];
  const float* b2 = (const float*)d_in[15];

  const int B = in_sizes[0] / (T_ * F_);  // 65536

  float* ws     = (float*)d_ws;
  float* stats0 = ws;                         // 32 floats
  float* stats3 = ws + 32;                    // 16 floats
  float* aff    = ws + 48;                    // 48 floats
  float* V1     = ws + 96;                    // 256 floats (W1+U1)
  float* hbuf   = ws + 352;                   // B*8
  float* cbuf   = hbuf + (size_t)B * 8;       // B*8
  float* bars   = cbuf + (size_t)B * 8;       // B*14*8

  const int gridB = B / 64;  // 128 threads = 4 waves x 16 batch rows

  k0_zero<<<1, 64, 0, stream>>>(ws, 48);
  k1_lstm0<<<gridB, 128, 0, stream>>>(X, W0, U0, b0, hbuf, cbuf, stats0);
  k2_fin0<<<1, 256, 0, stream>>>(stats0, gh, bh, gc, bc, W1, U1, aff, V1,
                                 1.0f / (float)B);
  k3_lstm1<<<gridB, 128, 0, stream>>>(hbuf, cbuf, aff, V1, b1, bars, stats3);
  k4_fin3<<<1, 32, 0, stream>>>(stats3, g3, b3, aff, 1.0f / ((float)B * RW));
  k5_lstm2<<<gridB, 128, 0, stream>>>(bars, aff, W2, U2, b2, (float*)d_out);
}